// ScaledDotProductAttention_5557687681284
// MI455X (gfx1250) — compile-verified
//
#include <hip/hip_runtime.h>
#include <math.h>

typedef float v2f __attribute__((ext_vector_type(2)));
typedef float v8f __attribute__((ext_vector_type(8)));

#define NB    8
#define NLQ   2048
#define NLK   2048
#define ND    64
#define QTILE 16
#define SROW  (NLK + 4)   // padded LDS row stride (floats): 2052 % 64 == 4 -> conflict-free columns

__global__ __launch_bounds__(128)
void sdpa_fp32_wmma_kernel(const float* __restrict__ q,
                           const float* __restrict__ k,
                           const float* __restrict__ v,
                           const float* __restrict__ scale_p,
                           const unsigned char* __restrict__ mask,
                           float* __restrict__ ctx,
                           float* __restrict__ attn)
{
    extern __shared__ float S[];              // [QTILE][SROW] = 128.25 KB (needs CDNA5 320KB WGP LDS)

    const int qt   = blockIdx.x;              // q tile: 0..127
    const int b    = blockIdx.y;              // batch:  0..7
    const int q0   = qt * QTILE;
    const int lane = threadIdx.x & 31;
    const int wave = threadIdx.x >> 5;        // 0..3
    const int h    = lane >> 4;               // K-half selector for A/B fragments
    const int ln   = lane & 15;               // M (A) / N (B) index
    const float scale = *scale_p;

    // ---------------- Phase 1: S = mask ? -inf : scale * (Q K^T) ----------------
    // Preload Q-tile A-fragments: 16 chunks of K=4.  A layout (16x4 f32):
    // lanes 0-15 hold {K=0,K=1} of row M=lane, lanes 16-31 hold {K=2,K=3}.
    v2f aQ[16];
    {
        const float* qrow = q + ((size_t)(b * NLQ + q0 + ln)) * ND;
        #pragma unroll
        for (int kk = 0; kk < 16; ++kk)
            aQ[kk] = *(const v2f*)(qrow + 4 * kk + 2 * h);
    }

    for (int kb = wave; kb < NLK / 16; kb += 4) {
        const float* krow = k + ((size_t)(b * NLK + kb * 16 + ln)) * ND;
        if (kb + 4 < NLK / 16) {
            // software prefetch of next assigned K block -> global_prefetch_b8
            __builtin_prefetch(k + ((size_t)(b * NLK + (kb + 4) * 16 + ln)) * ND, 0, 3);
        }
        v8f c = {0.f, 0.f, 0.f, 0.f, 0.f, 0.f, 0.f, 0.f};
        #pragma unroll
        for (int kk = 0; kk < 16; ++kk) {
            v2f bK = *(const v2f*)(krow + 4 * kk + 2 * h);   // B: K^T fragment
            c = __builtin_amdgcn_wmma_f32_16x16x4_f32(false, aQ[kk], false, bK,
                                                      (short)0, c, false, false);
        }
        // C layout: VGPR j holds row (j + 8*h), column = kb*16 + ln
        const int col = kb * 16 + ln;
        #pragma unroll
        for (int j = 0; j < 8; ++j) {
            const int m = j + 8 * h;
            const unsigned char mv = mask[((size_t)(b * NLQ + q0 + m)) * NLK + col];
            S[m * SROW + col] = mv ? -INFINITY : c[j] * scale;
        }
    }
    __syncthreads();

    // ---------------- Phase 2: row softmax in LDS; emit attention output ----------------
    for (int r = wave * 4; r < wave * 4 + 4; ++r) {
        float* srow = S + r * SROW;
        float mx = -INFINITY;
        for (int cidx = lane; cidx < NLK; cidx += 32)
            mx = fmaxf(mx, srow[cidx]);
        #pragma unroll
        for (int off = 16; off > 0; off >>= 1)
            mx = fmaxf(mx, __shfl_xor(mx, off, 32));

        float sum = 0.f;
        for (int cidx = lane; cidx < NLK; cidx += 32) {
            const float e = __expf(srow[cidx] - mx);
            srow[cidx] = e;
            sum += e;
        }
        #pragma unroll
        for (int off = 16; off > 0; off >>= 1)
            sum += __shfl_xor(sum, off, 32);

        const float inv = 1.0f / sum;
        float* arow = attn + ((size_t)(b * NLQ + q0 + r)) * NLK;
        for (int cidx = lane; cidx < NLK; cidx += 32) {
            const float p = srow[cidx] * inv;
            srow[cidx] = p;          // kept for P@V
            arow[cidx] = p;          // coalesced 128B/wave store
        }
    }
    __syncthreads();

    // ---------------- Phase 3: context = P @ V  (one 16x16 d-block per wave) ----------------
    {
        const int nb = wave;         // d-column block: columns nb*16 .. nb*16+15
        v8f c0 = {0.f, 0.f, 0.f, 0.f, 0.f, 0.f, 0.f, 0.f};
        v8f c1 = {0.f, 0.f, 0.f, 0.f, 0.f, 0.f, 0.f, 0.f};
        for (int kc = 0; kc < NLK / 4; kc += 2) {
            // A from LDS probabilities (8-byte aligned, conflict-free via SROW padding)
            v2f a0 = *(const v2f*)(S + ln * SROW + 4 * kc + 2 * h);
            v2f a1 = *(const v2f*)(S + ln * SROW + 4 * (kc + 1) + 2 * h);
            // B from V: rows 4*kc+2h, +1 ; coalesced across lanes 0-15 in d
            const int kr0 = 4 * kc + 2 * h;
            v2f b0, b1;
            b0.x = v[((size_t)(b * NLK + kr0))     * ND + nb * 16 + ln];
            b0.y = v[((size_t)(b * NLK + kr0 + 1)) * ND + nb * 16 + ln];
            b1.x = v[((size_t)(b * NLK + kr0 + 4)) * ND + nb * 16 + ln];
            b1.y = v[((size_t)(b * NLK + kr0 + 5)) * ND + nb * 16 + ln];
            c0 = __builtin_amdgcn_wmma_f32_16x16x4_f32(false, a0, false, b0,
                                                       (short)0, c0, false, false);
            c1 = __builtin_amdgcn_wmma_f32_16x16x4_f32(false, a1, false, b1,
                                                       (short)0, c1, false, false);
        }
        #pragma unroll
        for (int j = 0; j < 8; ++j) {
            const int m = j + 8 * h;
            ctx[((size_t)(b * NLQ + q0 + m)) * ND + nb * 16 + ln] = c0[j] + c1[j];
        }
    }
}

extern "C" void kernel_launch(void* const* d_in, const int* in_sizes, int n_in,
                              void* d_out, int out_size, void* d_ws, size_t ws_size,
                              hipStream_t stream) {
    (void)in_sizes; (void)n_in; (void)d_ws; (void)ws_size; (void)out_size;

    const float*         q     = (const float*)d_in[0];
    const float*         k     = (const float*)d_in[1];
    const float*         v     = (const float*)d_in[2];
    const float*         scale = (const float*)d_in[3];
    const unsigned char* mask  = (const unsigned char*)d_in[4];   // jax bool -> 1 byte

    float* ctx  = (float*)d_out;                                  // [8,2048,64]
    float* attn = (float*)d_out + (size_t)NB * NLQ * ND;          // [8,2048,2048]

    const size_t lds_bytes = (size_t)QTILE * SROW * sizeof(float); // 131328 B
    (void)hipFuncSetAttribute((const void*)sdpa_fp32_wmma_kernel,
                              hipFuncAttributeMaxDynamicSharedMemorySize,
                              (int)lds_bytes);

    dim3 grid(NLQ / QTILE, NB);   // 128 x 8 workgroups
    dim3 block(128);              // 4 wave32 waves
    sdpa_fp32_wmma_kernel<<<grid, block, lds_bytes, stream>>>(q, k, v, scale, mask, ctx, attn);
}